// GCNEncoder_1159641170174
// MI455X (gfx1250) — compile-verified
//
#include <hip/hip_runtime.h>
#include <hip/hip_bf16.h>

typedef __attribute__((ext_vector_type(2))) float v2f;
typedef __attribute__((ext_vector_type(8))) float v8f;

#define IN_CH 128
#define HID   15
#define HIDP  16
#define OUTC  32

// ---------------- degree / normalization ----------------

__global__ __launch_bounds__(256) void k_deg_init(unsigned int* __restrict__ deg, int N) {
  int i = blockIdx.x * blockDim.x + threadIdx.x;
  if (i < N) deg[i] = 1u;  // self-loop contributes 1 to every node's degree
}

__global__ __launch_bounds__(256) void k_deg_edges(const long long* __restrict__ ei,
                                                   unsigned int* __restrict__ deg, int E) {
  int i = blockIdx.x * blockDim.x + threadIdx.x;
  if (i < E) {
    int d = (int)ei[(long long)E + i];  // dst row of edge_index
    atomicAdd(&deg[d], 1u);
  }
}

__global__ __launch_bounds__(256) void k_dinv(const unsigned int* __restrict__ deg,
                                              float* __restrict__ dinv, int N) {
  int i = blockIdx.x * blockDim.x + threadIdx.x;
  if (i < N) {
    float d = (float)deg[i];
    dinv[i] = (d > 0.f) ? rsqrtf(d) : 0.f;
  }
}

// ---------------- layer 1 GEMM: h1 = x @ W1 (padded to 16 cols) ----------------
// One wave per 16-row tile; f32 WMMA 16x16x4, K-loop over 128.
// W1 is staged zero-padded + transposed in LDS so B-fragment loads are
// unconditional ds_load_b64 (no EXEC gymnastics in the inner loop).
// A layout (32-bit 16x4): lanes 0-15 hold M=lane%16, {K, K+1}; lanes 16-31 hold {K+2, K+3}.
// B layout mirrors C/D striping: VGPR g holds row K (lanes 0-15) / K+2 (lanes 16-31), N=lane%16.

__global__ __launch_bounds__(128) void k_gemm1(const float* __restrict__ x,
                                               const float* __restrict__ W1,
                                               float* __restrict__ h1, int N) {
  constexpr int WSTR = IN_CH + 2;          // 130: stride not a multiple of 64 banks
  __shared__ float Wt[16 * WSTR];          // Wt[n][k] = W1[k][n], col 15 zero-padded
  for (int i = threadIdx.x; i < 16 * IN_CH; i += 128) {
    int n = i >> 7;        // 0..15
    int k = i & (IN_CH - 1);
    Wt[n * WSTR + k] = (n < HID) ? W1[k * HID + n] : 0.f;
  }
  __syncthreads();

  const int lane = threadIdx.x & 31;
  const int tile = blockIdx.x * 4 + (threadIdx.x >> 5);
  const int row0 = tile * 16;
  if (row0 >= N) return;                   // wave-uniform: EXEC all-1s for WMMA
  const int r  = lane & 15;
  const int kh = (lane >> 4) * 2;          // 0 or 2
  int arow = row0 + r; if (arow >= N) arow = N - 1;
  const float* __restrict__ xr = x + (long long)arow * IN_CH + kh;
  const float* __restrict__ wr = &Wt[r * WSTR + kh];

  v8f c = {};
  #pragma unroll
  for (int k = 0; k < IN_CH; k += 4) {
    v2f a = *(const v2f*)(xr + k);         // global_load_b64 (8B aligned)
    v2f b = *(const v2f*)(wr + k);         // ds_load_b64     (8B aligned)
    c = __builtin_amdgcn_wmma_f32_16x16x4_f32(false, a, false, b, (short)0, c, false, false);
  }

  const int mb = (lane >> 4) * 8;
  #pragma unroll
  for (int g = 0; g < 8; ++g) {
    const int m = row0 + mb + g;
    if (m < N) h1[(long long)m * HIDP + r] = c[g];  // col 15 is exactly 0 (zero B pad)
  }
}

// ---------------- layer 1 aggregate ----------------

// agg1[n][c] = h1[n][c] * dinv[n]^2   (self-loop message; also zero-initializes buffer)
__global__ __launch_bounds__(256) void k_agg1_init(const float* __restrict__ h1,
                                                   const float* __restrict__ dinv,
                                                   float* __restrict__ agg1, int N) {
  int i = blockIdx.x * blockDim.x + threadIdx.x;
  if (i < N * HIDP) {
    float di = dinv[i >> 4];
    agg1[i] = h1[i] * di * di;
  }
}

// 16 lanes per edge: coalesced 64B gather of h1[src] row, coalesced f32 atomics to agg1[dst]
__global__ __launch_bounds__(256) void k_scatter1(const long long* __restrict__ ei,
                                                  const float* __restrict__ dinv,
                                                  const float* __restrict__ h1,
                                                  float* __restrict__ agg1, int E) {
  long long t = (long long)blockIdx.x * blockDim.x + threadIdx.x;
  int c = (int)(t & 15);
  long long e = t >> 4;
  if (e >= E) return;
  int s = (int)ei[e];
  int d = (int)ei[(long long)E + e];
  float nrm = dinv[s] * dinv[d];
  if (c < HID)
    atomicAdd(&agg1[(long long)d * HIDP + c], h1[(long long)s * HIDP + c] * nrm);
}

// ---------------- layer 2 GEMM: h3 = relu(agg1 + b1) @ W2 (bias+ReLU fused into A load) ----------------

__global__ __launch_bounds__(128) void k_gemm2(const float* __restrict__ agg1,
                                               const float* __restrict__ b1,
                                               const float* __restrict__ W2,
                                               float* __restrict__ h3, int N) {
  constexpr int WSTR2 = HIDP + 2;          // 18: bank-conflict-free stride
  __shared__ float Wt2[OUTC * WSTR2];      // Wt2[n][k] = W2[k][n], rows k>=15 zero-padded
  __shared__ float b1s[HIDP];              // b1 zero-padded to 16
  for (int i = threadIdx.x; i < OUTC * HIDP; i += 128) {
    int n = i >> 4;        // 0..31
    int k = i & 15;
    Wt2[n * WSTR2 + k] = (k < HID) ? W2[k * OUTC + n] : 0.f;
  }
  if (threadIdx.x < HIDP) b1s[threadIdx.x] = (threadIdx.x < HID) ? b1[threadIdx.x] : 0.f;
  __syncthreads();

  const int lane = threadIdx.x & 31;
  const int tile = blockIdx.x * 4 + (threadIdx.x >> 5);
  const int row0 = tile * 16;
  if (row0 >= N) return;
  const int r  = lane & 15;
  const int kh = (lane >> 4) * 2;
  int arow = row0 + r; if (arow >= N) arow = N - 1;
  const float* __restrict__ ar = agg1 + (long long)arow * HIDP + kh;

  v8f c0 = {}, c1 = {};
  #pragma unroll
  for (int k = 0; k < HIDP; k += 4) {
    const int kk = k + kh;
    v2f av = *(const v2f*)(ar + k);                       // global_load_b64
    v2f a;                                                // fused bias + ReLU
    a.x = fmaxf(av.x + b1s[kk],     0.f);                 // pad col: 0 + 0 -> 0
    a.y = fmaxf(av.y + b1s[kk + 1], 0.f);
    v2f b0  = *(const v2f*)&Wt2[r        * WSTR2 + kk];   // ds_load_b64, N-tile 0
    v2f b1v = *(const v2f*)&Wt2[(r + 16) * WSTR2 + kk];   // ds_load_b64, N-tile 1
    c0 = __builtin_amdgcn_wmma_f32_16x16x4_f32(false, a, false, b0,  (short)0, c0, false, false);
    c1 = __builtin_amdgcn_wmma_f32_16x16x4_f32(false, a, false, b1v, (short)0, c1, false, false);
  }

  const int mb = (lane >> 4) * 8;
  #pragma unroll
  for (int g = 0; g < 8; ++g) {
    const int m = row0 + mb + g;
    if (m < N) {
      h3[(long long)m * OUTC + r]      = c0[g];
      h3[(long long)m * OUTC + 16 + r] = c1[g];
    }
  }
}

// ---------------- layer 2 aggregate ----------------

// out[n][c] = h3[n][c]*dinv[n]^2 + b2[c]   (self-loop + bias; initializes d_out)
__global__ __launch_bounds__(256) void k_out_init(const float* __restrict__ h3,
                                                  const float* __restrict__ dinv,
                                                  const float* __restrict__ b2,
                                                  float* __restrict__ out, int N) {
  int i = blockIdx.x * blockDim.x + threadIdx.x;
  if (i < N * OUTC) {
    float di = dinv[i >> 5];
    out[i] = h3[i] * di * di + b2[i & 31];
  }
}

// full wave per edge: 128B coalesced gather + coalesced atomics
__global__ __launch_bounds__(256) void k_scatter2(const long long* __restrict__ ei,
                                                  const float* __restrict__ dinv,
                                                  const float* __restrict__ h3,
                                                  float* __restrict__ out, int E) {
  long long t = (long long)blockIdx.x * blockDim.x + threadIdx.x;
  int c = (int)(t & 31);
  long long e = t >> 5;
  if (e >= E) return;
  int s = (int)ei[e];
  int d = (int)ei[(long long)E + e];
  float nrm = dinv[s] * dinv[d];
  atomicAdd(&out[(long long)d * OUTC + c], h3[(long long)s * OUTC + c] * nrm);
}

// ---------------- launch ----------------

extern "C" void kernel_launch(void* const* d_in, const int* in_sizes, int n_in,
                              void* d_out, int out_size, void* d_ws, size_t ws_size,
                              hipStream_t stream) {
  const float*     x  = (const float*)d_in[0];
  const long long* ei = (const long long*)d_in[1];   // int64 edge_index, shape (2,E) flat
  const float*     W1 = (const float*)d_in[2];
  const float*     b1 = (const float*)d_in[3];
  const float*     W2 = (const float*)d_in[4];
  const float*     b2 = (const float*)d_in[5];
  float*           out = (float*)d_out;

  const int N = in_sizes[0] / IN_CH;   // 200000
  const int E = in_sizes[1] / 2;       // 6400000

  // workspace carve-up (all regions fully written before being read)
  char* ws = (char*)d_ws;
  size_t off = 0;
  unsigned int* deg  = (unsigned int*)(ws + off); off += (size_t)N * sizeof(unsigned int);
  float*        dinv = (float*)(ws + off);        off += (size_t)N * sizeof(float);
  float*        h1   = (float*)(ws + off);        off += (size_t)N * HIDP * sizeof(float);
  float*        agg1 = (float*)(ws + off);        off += (size_t)N * HIDP * sizeof(float);
  float*        h3   = (float*)(ws + off);        off += (size_t)N * OUTC * sizeof(float);
  (void)ws_size; (void)n_in; (void)out_size;

  const int B = 256;
  const int tiles = (N + 15) / 16;
  const int gblk  = (tiles + 3) / 4;               // 4 waves (tiles) per 128-thread block

  k_deg_init <<<(N + B - 1) / B, B, 0, stream>>>(deg, N);
  k_deg_edges<<<(E + B - 1) / B, B, 0, stream>>>(ei, deg, E);
  k_dinv     <<<(N + B - 1) / B, B, 0, stream>>>(deg, dinv, N);

  k_gemm1    <<<gblk, 128, 0, stream>>>(x, W1, h1, N);
  k_agg1_init<<<((size_t)N * HIDP + B - 1) / B, B, 0, stream>>>(h1, dinv, agg1, N);
  {
    long long th = (long long)E * 16;
    k_scatter1<<<(unsigned int)((th + B - 1) / B), B, 0, stream>>>(ei, dinv, h1, agg1, E);
  }

  k_gemm2    <<<gblk, 128, 0, stream>>>(agg1, b1, W2, h3, N);
  k_out_init <<<((size_t)N * OUTC + B - 1) / B, B, 0, stream>>>(h3, dinv, b2, out, N);
  {
    long long th = (long long)E * 32;
    k_scatter2<<<(unsigned int)((th + B - 1) / B), B, 0, stream>>>(ei, dinv, h3, out, E);
  }
}